// QuantumLayer_62886911148550
// MI455X (gfx1250) — compile-verified
//
#include <hip/hip_runtime.h>

typedef __attribute__((ext_vector_type(2))) float v2f;
typedef __attribute__((ext_vector_type(8))) float v8f;

#define DIM 256          // 2^8 amplitudes
#define KDIM 512         // real-embedded inner dimension
#define SAMP_PER_WG 64   // samples per workgroup (4 waves x 16 rows)
#define THREADS2 128
#define PHI_STR 516      // padded row stride (floats) to avoid LDS bank conflicts

// ---------------- Kernel 1: build the fixed 512x512 real operator W ----------------
// Column s of U is obtained by applying the gate sequence to basis state e_s.
// All 256 columns are independent; one block per column, one thread per amplitude.

__device__ __forceinline__ void cnot_g(float* sr, float* si, int r, int c, int t) {
  const int mc = 1 << (7 - c), mt = 1 << (7 - t);
  const int src = (r & mc) ? (r ^ mt) : r;
  const float ar = sr[src], ai = si[src];
  __syncthreads();
  sr[r] = ar; si[r] = ai;
  __syncthreads();
}

__device__ __forceinline__ void ry_g(float* sr, float* si, int r, int w, float cs, float sn) {
  const int m = 1 << (7 - w);
  const int p = r ^ m;
  const float sg = (r & m) ? sn : -sn;   // a0' = c*a0 - s*a1 ; a1' = s*a0 + c*a1
  const float nr = cs * sr[r] + sg * sr[p];
  const float ni = cs * si[r] + sg * si[p];
  __syncthreads();
  sr[r] = nr; si[r] = ni;
  __syncthreads();
}

__device__ __forceinline__ void rx_g(float* sr, float* si, int r, int w, float cs, float sn) {
  const int m = 1 << (7 - w);
  const int p = r ^ m;                    // a' = c*a - i*s*a_partner (same form for both halves)
  const float nr = cs * sr[r] + sn * si[p];
  const float ni = cs * si[r] - sn * sr[p];
  __syncthreads();
  sr[r] = nr; si[r] = ni;
  __syncthreads();
}

__global__ __launch_bounds__(256) void build_W_kernel(const float* __restrict__ qp,
                                                      float* __restrict__ Wg) {
  __shared__ float sr[DIM];
  __shared__ float si[DIM];
  const int r = threadIdx.x;   // amplitude index within the column
  const int s = blockIdx.x;    // which basis column
  float pc[7], ps[7];
  #pragma unroll
  for (int i = 0; i < 7; ++i) sincosf(qp[i], &ps[i], &pc[i]);

  sr[r] = (r == s) ? 1.0f : 0.0f;
  si[r] = 0.0f;
  __syncthreads();

  // CNOT chain 0->1 ... 6->7
  #pragma unroll
  for (int i = 0; i < 7; ++i) cnot_g(sr, si, r, i, i + 1);
  // RY on wires 0,2,4,6
  ry_g(sr, si, r, 0, pc[0], ps[0]);
  ry_g(sr, si, r, 2, pc[1], ps[1]);
  ry_g(sr, si, r, 4, pc[2], ps[2]);
  ry_g(sr, si, r, 6, pc[3], ps[3]);
  cnot_g(sr, si, r, 7, 0);
  // RX on wires 1,3,5
  rx_g(sr, si, r, 1, pc[4], ps[4]);
  rx_g(sr, si, r, 3, pc[5], ps[5]);
  rx_g(sr, si, r, 5, pc[6], ps[6]);
  cnot_g(sr, si, r, 0, 1);
  cnot_g(sr, si, r, 2, 3);
  cnot_g(sr, si, r, 4, 5);
  cnot_g(sr, si, r, 6, 7);
  cnot_g(sr, si, r, 1, 3);
  cnot_g(sr, si, r, 3, 5);
  cnot_g(sr, si, r, 5, 7);

  // Real embedding: Psi = A_rows @ W, W[t][n]:
  //   W[s][n]       =  Re U[n,s]    W[s][256+n]     = Im U[n,s]
  //   W[s+256][n]   = -Im U[n,s]    W[s+256][256+n] = Re U[n,s]
  Wg[s * KDIM + r]                = sr[r];
  Wg[s * KDIM + DIM + r]          = si[r];
  Wg[(s + DIM) * KDIM + r]        = -si[r];
  Wg[(s + DIM) * KDIM + DIM + r]  = sr[r];
}

// ---------------- Kernel 2: Phi build + fused GEMM/quadratic-form via WMMA ----------------

__global__ __launch_bounds__(THREADS2) void qform_kernel(const float* __restrict__ x,
                                                         const float* __restrict__ Wg,
                                                         float* __restrict__ out, int B) {
  __shared__ float phi[SAMP_PER_WG * PHI_STR]; // 64 rows x (256 re | 256 im | pad)
  __shared__ float wt[KDIM * 16];              // staged W tile: 512 K x 16 N
  const int tid  = threadIdx.x;
  const int lane = tid & 31;
  const int wave = tid >> 5;
  const int l16  = lane & 15;
  const int half = lane >> 4;
  const int b0   = blockIdx.x * SAMP_PER_WG;

  // Build encoded phase vectors: phi_b[s] = (1/16) * prod_i e^{+-i x[b,i]} (qubit 0 = MSB)
  if (tid < SAMP_PER_WG) {
    const int b = b0 + tid;
    float* pr = &phi[tid * PHI_STR];
    float* pi = pr + DIM;
    if (b < B) {
      float cs[8], sn[8];
      #pragma unroll
      for (int i = 0; i < 8; ++i) sincosf(x[b * 8 + i], &sn[i], &cs[i]);
      pr[0] = 0.0625f;   // 2^-4 normalization baked in (squares give 1/256)
      pi[0] = 0.0f;
      int len = 1;
      #pragma unroll
      for (int i = 0; i < 8; ++i) {       // Kronecker doubling, MSB-first
        for (int j = len - 1; j >= 0; --j) {
          const float orr = pr[j], oii = pi[j];
          pr[2 * j + 1] = orr * cs[i] - oii * sn[i];   // bit=1: * e^{+ia}
          pi[2 * j + 1] = orr * sn[i] + oii * cs[i];
          pr[2 * j]     = orr * cs[i] + oii * sn[i];   // bit=0: * e^{-ia}
          pi[2 * j]     = oii * cs[i] - orr * sn[i];
        }
        len <<= 1;
      }
    } else {
      for (int j = 0; j < KDIM; ++j) pr[j] = 0.0f;     // zero tail rows
    }
  }

  float acc[8];
  #pragma unroll
  for (int j = 0; j < 8; ++j) acc[j] = 0.0f;

  const int m = wave * 16 + l16;                 // this lane's A-row (sample within WG)
  const float* prow = &phi[m * PHI_STR];

  for (int n0 = 0; n0 < KDIM; n0 += 16) {
    __syncthreads();
    // Cooperative stage of W[:, n0:n0+16] into LDS (shared by all 4 waves)
    for (int idx = tid; idx < KDIM * 16; idx += THREADS2)
      wt[idx] = Wg[(idx >> 4) * KDIM + n0 + (idx & 15)];
    __syncthreads();

    v8f c = {};
    #pragma unroll 8
    for (int k = 0; k < KDIM; k += 4) {
      const int kk = k + 2 * half;               // lanes 0-15: K=k,k+1 ; lanes 16-31: K=k+2,k+3
      v2f a, bf;
      a.x  = prow[kk];
      a.y  = prow[kk + 1];
      bf.x = wt[kk * 16 + l16];
      bf.y = wt[(kk + 1) * 16 + l16];
      c = __builtin_amdgcn_wmma_f32_16x16x4_f32(false, a, false, bf, (short)0, c,
                                                false, false);
    }
    // out_b += z(n) * Psi[b,n]^2 ; z = -1 when amplitude-index bit7 (qubit 0) is set
    const float sgn = ((n0 + l16) & 128) ? -1.0f : 1.0f;
    #pragma unroll
    for (int j = 0; j < 8; ++j) acc[j] += sgn * c[j] * c[j];
  }

  // Reduce the 16 N-lanes of each half (rows j+8*half live in VGPR j)
  #pragma unroll
  for (int j = 0; j < 8; ++j) {
    acc[j] += __shfl_xor(acc[j], 1, 32);
    acc[j] += __shfl_xor(acc[j], 2, 32);
    acc[j] += __shfl_xor(acc[j], 4, 32);
    acc[j] += __shfl_xor(acc[j], 8, 32);
  }
  const int rowbase = b0 + wave * 16 + 8 * half;
  #pragma unroll
  for (int j = 0; j < 8; ++j) {
    const int ob = rowbase + j;
    if (l16 == j && ob < B) out[ob] = acc[j];
  }
}

extern "C" void kernel_launch(void* const* d_in, const int* in_sizes, int n_in,
                              void* d_out, int out_size, void* d_ws, size_t ws_size,
                              hipStream_t stream) {
  const float* x  = (const float*)d_in[0];   // [B, 8] f32
  const float* qp = (const float*)d_in[1];   // [7] f32
  float* out = (float*)d_out;                // [B] f32
  float* Wg  = (float*)d_ws;                 // 512*512*4 = 1 MB scratch

  build_W_kernel<<<DIM, DIM, 0, stream>>>(qp, Wg);

  const int B = in_sizes[0] / 8;
  const int nwg = (B + SAMP_PER_WG - 1) / SAMP_PER_WG;
  qform_kernel<<<nwg, THREADS2, 0, stream>>>(x, Wg, out, B);
}